// GCN_1_28922309771670
// MI455X (gfx1250) — compile-verified
//
#include <hip/hip_runtime.h>
#include <hip/hip_bf16.h>

typedef __attribute__((ext_vector_type(2))) float v2f;
typedef __attribute__((ext_vector_type(8))) float v8f;

#define NN 50000
#define NE 800000
#define NG 64

// ---------------- degree / norm ----------------
__global__ void k_deg_init(float* __restrict__ deg, int n) {
    int i = blockIdx.x * blockDim.x + threadIdx.x;
    if (i < n) deg[i] = 1.0f;                       // self loop
}
__global__ void k_deg_count(const int* __restrict__ dst, float* __restrict__ deg, int e) {
    int i = blockIdx.x * blockDim.x + threadIdx.x;
    if (i < e) atomicAdd(&deg[dst[i]], 1.0f);
}
__global__ void k_rsqrt(float* __restrict__ deg, int n) {
    int i = blockIdx.x * blockDim.x + threadIdx.x;
    if (i < n) deg[i] = rsqrtf(deg[i]);
}

// ---------------- layer-1 GEMM (K = 9, Nc = 128) ----------------
__global__ void k_gemm9(const float* __restrict__ X, const float* __restrict__ W,
                        float* __restrict__ C, int M) {
    int i = blockIdx.x * blockDim.x + threadIdx.x;
    if (i >= M * 128) return;
    int m = i >> 7, n = i & 127;
    const float* xr = X + m * 9;
    float s = 0.f;
#pragma unroll
    for (int k = 0; k < 9; ++k) s = fmaf(xr[k], W[k * 128 + n], s);
    C[i] = s;
}

// ---------------- weight transpose: Wt[n][k] = W[k][n] (tiny, once per layer) ----------------
__global__ void k_transpose(const float* __restrict__ W, float* __restrict__ Wt,
                            int K, int Nc) {
    int i = blockIdx.x * blockDim.x + threadIdx.x;
    if (i >= K * Nc) return;
    int k = i / Nc, n = i - k * Nc;
    Wt[(size_t)n * K + k] = W[i];
}

// ---------------- fp32 WMMA GEMM: C[M x 256] = A[M x K] @ B[K x 256], B transposed ----
// One wave -> two 16x16 tiles (cols n0 and n0+128) via V_WMMA_F32_16X16X4_F32.
// A layout (ISA 7.12.2, 32-bit A 16x4): lane&15 = M row, lane>>4 selects k-pair {0,1}/{2,3}.
// B from Bt (col-major): per lane a contiguous float2 Bt[col][k+2hi .. k+2hi+1].
// One A b64 load feeds two wmma ops; ReLU fused as a single v_med3 per element.
template <bool RELU>
__global__ void __launch_bounds__(256)
k_gemm_wmma2(const float* __restrict__ A, const float* __restrict__ Bt,
             float* __restrict__ C, int M, int K) {
    const int Nc = 256;
    const int lane = threadIdx.x & 31;
    const int wave = threadIdx.x >> 5;
    const int m0 = blockIdx.x * 16;
    const int n0 = wave * 16;                       // 8 waves cover cols 0..127
    const int ml = lane & 15;
    const int hi = lane >> 4;
    const int col = n0 + ml;

    const float2* ap  = (const float2*)(A + (size_t)(m0 + ml) * K + 2 * hi);
    const float2* bp0 = (const float2*)(Bt + (size_t)col * K + 2 * hi);
    const float2* bp1 = (const float2*)(Bt + (size_t)(col + 128) * K + 2 * hi);

    v8f acc0 = {}, acc1 = {};
    for (int k = 0; k < K; k += 4) {
        float2 av = *ap;  ap  += 2;                 // advance 4 floats
        float2 b0 = *bp0; bp0 += 2;
        float2 b1 = *bp1; bp1 += 2;
        if (RELU) {
            av.x = __builtin_amdgcn_fmed3f(av.x, 0.f, __builtin_inff());
            av.y = __builtin_amdgcn_fmed3f(av.y, 0.f, __builtin_inff());
        }
        v2f a;  a.x = av.x;  a.y = av.y;
        v2f v0; v0.x = b0.x; v0.y = b0.y;
        v2f v1; v1.x = b1.x; v1.y = b1.y;
        acc0 = __builtin_amdgcn_wmma_f32_16x16x4_f32(
            false, a, false, v0, (short)0, acc0, false, false);
        acc1 = __builtin_amdgcn_wmma_f32_16x16x4_f32(
            false, a, false, v1, (short)0, acc1, false, false);
    }
    // D layout: vgpr r -> row m0 + r + hi*8
#pragma unroll
    for (int r = 0; r < 8; ++r) {
        int orow = m0 + r + hi * 8;
        if (orow < M) {
            C[(size_t)orow * Nc + col]       = acc0[r];
            C[(size_t)orow * Nc + col + 128] = acc1[r];
        }
    }
}

// ---------------- scatter: self-loop + bias init (no atomics) ----------------
__global__ void k_self_bias(const float* __restrict__ tmp, const float* __restrict__ dinv,
                            const float* __restrict__ bias, float* __restrict__ out,
                            int n, int F, int logF) {
    int i = blockIdx.x * blockDim.x + threadIdx.x;
    if (i >= n * F) return;
    int node = i >> logF;
    int f = i & (F - 1);
    float d = dinv[node];
    out[i] = fmaf(tmp[i], d * d, bias[f]);
}

// ---------------- scatter: edge atomic add (float4 gather) ----------------
__global__ void k_edge_scatter(const int* __restrict__ src, const int* __restrict__ dst,
                               const float* __restrict__ dinv, const float* __restrict__ tmp,
                               float* __restrict__ out, int e, int F) {
    int eid = blockIdx.x * blockDim.x + threadIdx.x;
    if (eid >= e) return;
    int fq = blockIdx.y;                            // 0 .. F/4-1
    int s = src[eid], d = dst[eid];
    float w = dinv[s] * dinv[d];
    float4 v = reinterpret_cast<const float4*>(tmp + (size_t)s * F)[fq];
    float* o = out + (size_t)d * F + fq * 4;
    atomicAdd(o + 0, v.x * w);
    atomicAdd(o + 1, v.y * w);
    atomicAdd(o + 2, v.z * w);
    atomicAdd(o + 3, v.w * w);
}

// ---------------- global max pool (monotonic uint encoding) ----------------
__device__ __forceinline__ unsigned f2key(float f) {
    unsigned u = __float_as_uint(f);
    return (u & 0x80000000u) ? ~u : (u | 0x80000000u);
}
__device__ __forceinline__ float key2f(unsigned k) {
    unsigned u = (k & 0x80000000u) ? (k ^ 0x80000000u) : ~k;
    return __uint_as_float(u);
}
__global__ void k_zero_u32(unsigned* __restrict__ p, int n) {
    int i = blockIdx.x * blockDim.x + threadIdx.x;
    if (i < n) p[i] = 0u;
}
__global__ void k_pool_max(const float* __restrict__ h, const int* __restrict__ batch,
                           unsigned* __restrict__ keys, int n) {
    const int CH = 16;                               // consecutive nodes per thread
    const int F = 256;
    int nch = (n + CH - 1) / CH;
    int i = blockIdx.x * blockDim.x + threadIdx.x;
    if (i >= nch * F) return;
    int f = i & 255;
    int c = i >> 8;
    int n0 = c * CH;
    int n1 = min(n0 + CH, n);
    int curb = batch[n0];
    unsigned best = 0u;
    for (int nd = n0; nd < n1; ++nd) {
        int bb = batch[nd];
        if (bb != curb) {
            atomicMax(&keys[(size_t)curb * F + f], best);
            curb = bb; best = 0u;
        }
        unsigned k = f2key(h[(size_t)nd * F + f]);
        best = max(best, k);
    }
    atomicMax(&keys[(size_t)curb * F + f], best);
}
__global__ void k_decode(const unsigned* __restrict__ keys, float* __restrict__ g, int n) {
    int i = blockIdx.x * blockDim.x + threadIdx.x;
    if (i < n) g[i] = key2f(keys[i]);
}

// ---------------- dense head (M = 64, tiny) ----------------
__global__ void k_dense(const float* __restrict__ A, const float* __restrict__ W,
                        const float* __restrict__ bias, float* __restrict__ out,
                        int M, int K, int Nc, int relu) {
    int i = blockIdx.x * blockDim.x + threadIdx.x;
    if (i >= M * Nc) return;
    int m = i / Nc, n = i - m * Nc;
    float s = bias[n];
    const float* a = A + (size_t)m * K;
    for (int k = 0; k < K; ++k) s = fmaf(a[k], W[(size_t)k * Nc + n], s);
    if (relu) s = fmaxf(s, 0.f);
    out[i] = s;
}

extern "C" void kernel_launch(void* const* d_in, const int* in_sizes, int n_in,
                              void* d_out, int out_size, void* d_ws, size_t ws_size,
                              hipStream_t stream) {
    const float* x    = (const float*)d_in[0];
    const int*   ei   = (const int*)d_in[1];        // [2, E]: row0 src, row1 dst
    const int*   bat  = (const int*)d_in[2];
    const float* W1   = (const float*)d_in[3];  const float* b1 = (const float*)d_in[4];
    const float* W2   = (const float*)d_in[5];  const float* b2 = (const float*)d_in[6];
    const float* W3   = (const float*)d_in[7];  const float* b3 = (const float*)d_in[8];
    const float* W4   = (const float*)d_in[9];  const float* b4 = (const float*)d_in[10];
    const float* Wl2  = (const float*)d_in[11]; const float* bl2 = (const float*)d_in[12];
    const float* Wl3  = (const float*)d_in[13]; const float* bl3 = (const float*)d_in[14];
    const float* Wl   = (const float*)d_in[15]; const float* bl  = (const float*)d_in[16];
    const int* srcI = ei;
    const int* dstI = ei + NE;

    // workspace carve-up
    float* dinv = (float*)d_ws;                              // NN
    float* bufA = dinv + ((NN + 255) & ~255);                // NN*256 (gemm out)
    float* bufB = bufA + (size_t)NN * 256;                   // NN*256 (conv out)
    unsigned* keys = (unsigned*)(bufB + (size_t)NN * 256);   // NG*256
    float* gdec = (float*)(keys + NG * 256);                 // NG*256
    float* hh1 = gdec + NG * 256;                            // NG*128
    float* hh2 = hh1 + NG * 128;                             // NG*128
    float* Wt  = hh2 + NG * 128;                             // 256*256 (transposed weights)

    const int T = 256;
    // 1) deg -> dinv
    k_deg_init<<<(NN + T - 1) / T, T, 0, stream>>>(dinv, NN);
    k_deg_count<<<(NE + T - 1) / T, T, 0, stream>>>(dstI, dinv, NE);
    k_rsqrt<<<(NN + T - 1) / T, T, 0, stream>>>(dinv, NN);

    // 2) layer 1: x @ W1 (K=9) -> bufA; scatter -> bufB (F=128)
    k_gemm9<<<((size_t)NN * 128 + T - 1) / T, T, 0, stream>>>(x, W1, bufA, NN);
    k_self_bias<<<((size_t)NN * 128 + T - 1) / T, T, 0, stream>>>(bufA, dinv, b1, bufB, NN, 128, 7);
    {
        dim3 g((NE + T - 1) / T, 128 / 4);
        k_edge_scatter<<<g, T, 0, stream>>>(srcI, dstI, dinv, bufA, bufB, NE, 128);
    }

    // layers 2..4: transpose W, WMMA GEMM (relu fused on A-load), scatter; ping-pong B->A->B
    const float* Ws[3] = { W2, W3, W4 };
    const float* bs[3] = { b2, b3, b4 };
    int Ks[3] = { 128, 256, 256 };
    for (int l = 0; l < 3; ++l) {
        int K = Ks[l];
        k_transpose<<<(K * 256 + T - 1) / T, T, 0, stream>>>(Ws[l], Wt, K, 256);
        k_gemm_wmma2<true><<<NN / 16, 256, 0, stream>>>(bufB, Wt, bufA, NN, K);
        k_self_bias<<<((size_t)NN * 256 + T - 1) / T, T, 0, stream>>>(bufA, dinv, bs[l], bufB, NN, 256, 8);
        dim3 ge((NE + T - 1) / T, 256 / 4);
        k_edge_scatter<<<ge, T, 0, stream>>>(srcI, dstI, dinv, bufA, bufB, NE, 256);
    }

    // 3) global max pool over batch -> gdec [64, 256]
    k_zero_u32<<<(NG * 256 + T - 1) / T, T, 0, stream>>>(keys, NG * 256);
    {
        int nch = (NN + 15) / 16;
        k_pool_max<<<((size_t)nch * 256 + T - 1) / T, T, 0, stream>>>(bufB, bat, keys, NN);
    }
    k_decode<<<(NG * 256 + T - 1) / T, T, 0, stream>>>(keys, gdec, NG * 256);

    // 4) dense head
    k_dense<<<(NG * 128 + T - 1) / T, T, 0, stream>>>(gdec, Wl2, bl2, hh1, NG, 256, 128, 1);
    k_dense<<<(NG * 128 + T - 1) / T, T, 0, stream>>>(hh1, Wl3, bl3, hh2, NG, 128, 128, 1);
    k_dense<<<(NG * 10 + T - 1) / T, T, 0, stream>>>(hh2, Wl, bl, (float*)d_out, NG, 128, 10, 0);
}